// GaussianVectorQuantizer_85358180041232
// MI455X (gfx1250) — compile-verified
//
#include <hip/hip_runtime.h>
#include <hip/hip_bf16.h>

// ---------------------------------------------------------------------------
// Gaussian Vector Quantizer (train path), fused for MI455X (gfx1250).
//   N=65536 tokens, D=64, K=512.  Both GEMMs via V_WMMA_F32_16X16X4_F32.
//   Codebook staged into LDS by the Tensor Data Mover (padded rows via
//   D# pad_enable), logits live only in LDS.
// ---------------------------------------------------------------------------

typedef __attribute__((ext_vector_type(2))) float    v2f;
typedef __attribute__((ext_vector_type(8))) float    v8f;
typedef __attribute__((ext_vector_type(4))) unsigned v4u;
typedef __attribute__((ext_vector_type(8))) int      v8i;
typedef __attribute__((ext_vector_type(4))) int      v4i;

#if defined(__has_builtin)
#if __has_builtin(__builtin_amdgcn_tensor_load_to_lds) && \
    __has_builtin(__builtin_amdgcn_s_wait_tensorcnt)
#define USE_TDM 1
#endif
#endif
#ifndef USE_TDM
#define USE_TDM 0
#endif

__device__ __forceinline__ v8f wmma_f32_4(v2f a, v2f b, v8f c) {
  // 8 args: (neg_a, A, neg_b, B, c_mod, C, reuse_a, reuse_b)
  return __builtin_amdgcn_wmma_f32_16x16x4_f32(false, a, false, b, (short)0, c,
                                               false, false);
}

// --- exact JAX threefry2x32, key = jax.random.key(42) -> (0, 42) ------------
// For shape (2^25,) jax splits counts [0,2^25) into halves (x0,x1) pairwise.
__device__ __forceinline__ float gumbel_from_index(unsigned i) {
  unsigned j  = i & 0x00FFFFFFu;
  unsigned x0 = j, x1 = j | 0x01000000u;
  const unsigned k0 = 0u, k1 = 42u;
  const unsigned k2 = k0 ^ k1 ^ 0x1BD11BDAu;
  x0 += k0; x1 += k1;
#define TF_R(r) { x0 += x1; x1 = (x1 << (r)) | (x1 >> (32 - (r))); x1 ^= x0; }
  TF_R(13) TF_R(15) TF_R(26) TF_R(6)   x0 += k1; x1 += k2 + 1u;
  TF_R(17) TF_R(29) TF_R(16) TF_R(24)  x0 += k2; x1 += k0 + 2u;
  TF_R(13) TF_R(15) TF_R(26) TF_R(6)   x0 += k0; x1 += k1 + 3u;
  TF_R(17) TF_R(29) TF_R(16) TF_R(24)  x0 += k1; x1 += k2 + 4u;
  TF_R(13) TF_R(15) TF_R(26) TF_R(6)   x0 += k2; x1 += k0 + 5u;
#undef TF_R
  unsigned bits = (i < 0x01000000u) ? x0 : x1;
  unsigned fb = (bits >> 9) | 0x3F800000u;
  float U = __uint_as_float(fb) - 1.0f;          // jax uniform in [0,1)
  return -__logf(-__logf(U + 1e-10f) + 1e-10f);  // gumbel
}

// --- LDS layout (floats), dynamic shared = 288,800 B (<= 320 KB/WGP) --------
#define CB_OFF   0        // codebook, 512 rows x stride 66  (pad vs banks)
#define CB_STR   66
#define CN_OFF   33792    // ||c||^2, 512
#define S_OFF    34304    // s / enc, 64 rows x stride 516
#define S_STR    516
#define ZQ_OFF   67328    // zq transpose tile, 64 d x stride 68
#define ZQ_STR   68
#define AVG_OFF  71680    // per-block avg_probs accumulator, 512
#define RED_OFF  72192    // scalars
#define SM_FLOATS 72200

extern "C" __global__ void __launch_bounds__(128)
vq_main_kernel(const float* __restrict__ z, const float* __restrict__ var_q,
               const float* __restrict__ cb, float* __restrict__ out,
               float* __restrict__ ws) {
  extern __shared__ float sm[];
  float* CB  = sm + CB_OFF;
  float* CN  = sm + CN_OFF;
  float* S   = sm + S_OFF;
  float* ZQ  = sm + ZQ_OFF;
  float* AVG = sm + AVG_OFF;
  float* RED = sm + RED_OFF;

  const int tid  = threadIdx.x;
  const int wave = tid >> 5;
  const int lane = tid & 31;
  const int lo16 = lane & 15;
  const int hi   = lane >> 4;        // 0 or 1
  const int blockBase = blockIdx.x * 64;

  // ---- stage codebook into LDS (row stride 64 + 2 pad floats) ----
#if USE_TDM
  if (wave == 0) {
    // Tensor DMA descriptor: 2D 512x64 f32 tile, LDS pad 2 DWORDs / 64 DWORDs
    const unsigned ldsaddr = (unsigned)(unsigned long long)(CB); // LDS offset
    const unsigned long long ga = (unsigned long long)cb;
    v4u g0;
    g0.x = 1u;                                     // count=1, user descriptor
    g0.y = ldsaddr;                                // lds_addr [63:32]
    g0.z = (unsigned)(ga & 0xFFFFFFFFull);         // global_addr low
    g0.w = (unsigned)((ga >> 32) & 0x01FFFFFFull)  // global_addr [56:32]
           | 0x80000000u;                          // type=2 ("image")
    v8i g1;
    g1[0] = (int)((2u << 16)      // data_size = 4B
                | (1u << 20)      // pad_enable
                | (5u << 22)      // pad_interval: 2^(5+1)=64 DWORDs
                | (1u << 25));    // pad_amount: 2 DWORDs
    g1[1] = (int)(64u << 16);     // tensor_dim0 = 64   ([79:48] low part)
    g1[2] = (int)(512u << 16);    // tensor_dim1 = 512  ([111:80] low part)
    g1[3] = (int)(64u << 16);     // tile_dim0 = 64     ([127:112])
    g1[4] = 512;                  // tile_dim1 = 512    ([143:128])
    g1[5] = 64;                   // tensor_dim0_stride = 64 ([207:160] low)
    g1[6] = 0;
    g1[7] = 0;
    v4i gz = {0, 0, 0, 0};
#if __clang_major__ >= 23
    v8i gz8 = {0, 0, 0, 0, 0, 0, 0, 0};
    __builtin_amdgcn_tensor_load_to_lds(g0, g1, gz, gz, gz8, 0);
#else
    __builtin_amdgcn_tensor_load_to_lds(g0, g1, gz, gz, 0);
#endif
    __builtin_amdgcn_s_wait_tensorcnt(0);
  }
#else
  {
    const float4* src = (const float4*)cb;     // 8192 float4's
    for (int i = 0; i < 64; ++i) {
      float4 v = src[i * 128 + tid];
      int e = (i * 128 + tid) * 4;
      int n = e >> 6, d = e & 63;
      float* p = CB + n * CB_STR + d;
      p[0] = v.x; p[1] = v.y; p[2] = v.z; p[3] = v.w;
    }
  }
#endif
  for (int i = tid; i < 512; i += 128) AVG[i] = 0.0f;
  if (tid == 0) RED[0] = 0.0f;
  __syncthreads();

  // ---- codebook norms (4 codes / thread) ----
  for (int q = 0; q < 4; ++q) {
    int n = tid * 4 + q;
    const float* row = CB + n * CB_STR;
    float acc = 0.0f;
#pragma unroll
    for (int d = 0; d < 64; ++d) acc += row[d] * row[d];
    CN[n] = acc;
  }
  __syncthreads();

  const float w = 0.5f / fmaxf(var_q[0], 1e-10f);

  // ================= GEMM1: s = 2w * (z @ cb^T) - w*||c||^2 =================
  // A (16x4 f32 WMMA layout): lane holds token = lo16, dims 4k+2*hi+{0,1}
  const int tok = blockBase + wave * 16 + lo16;
  const int b64 = (tok >> 10) * 64;
  const int pix = tok & 1023;
  v2f a[16];
#pragma unroll
  for (int k0 = 0; k0 < 16; ++k0) {
    int d0 = 4 * k0 + 2 * hi;
    a[k0].x = z[((b64 + d0) << 10) + pix];
    a[k0].y = z[((b64 + d0 + 1) << 10) + pix];
  }

  for (int nt = 0; nt < 32; ++nt) {
    const int n0 = nt * 16;
    const float cn = CN[n0 + lo16];
    const float* cbrow = CB + (n0 + lo16) * CB_STR + 2 * hi;
    v8f c0 = {}, c1 = {};                  // dual chains for WMMA ILP
#pragma unroll
    for (int k0 = 0; k0 < 16; k0 += 2) {
      v2f bv0 = *(const v2f*)(cbrow + 4 * k0);       // cb[n][4k+2hi .. +1]
      v2f bv1 = *(const v2f*)(cbrow + 4 * k0 + 4);
      c0 = wmma_f32_4(a[k0], bv0, c0);
      c1 = wmma_f32_4(a[k0 + 1], bv1, c1);
    }
    const float negwc = -w * cn;
    const int rbase = wave * 16 + 8 * hi;
#pragma unroll
    for (int vg = 0; vg < 8; ++vg) {
      float s = 2.0f * w * (c0[vg] + c1[vg]) + negwc;
      S[(rbase + vg) * S_STR + n0 + lo16] = s;
    }
  }
  __syncthreads();

  // ================= softmax + gumbel-softmax (2 threads / token) ==========
  const int mloc  = tid >> 1;
  const int half  = tid & 1;
  const int token = blockBase + mloc;
  const float* Srow = S + mloc * S_STR + half * 256;

  // pass A: max over s; online (max,sum) for gumbel logits u = 2*(s+g)
  float m1 = -INFINITY, m2 = -INFINITY, Z2 = 0.0f;
  for (int i = 0; i < 256; ++i) {
    float s = Srow[i];
    m1 = fmaxf(m1, s);
    float g = gumbel_from_index((unsigned)(token * 512 + half * 256 + i));
    float u = 2.0f * (s + g);
    if (u > m2) { Z2 = Z2 * __expf(m2 - u) + 1.0f; m2 = u; }
    else        { Z2 += __expf(u - m2); }
  }
  // combine the two halves of each token (adjacent lanes, same wave32)
  m1 = fmaxf(m1, __shfl_xor(m1, 1, 32));
  {
    float m2o = __shfl_xor(m2, 1, 32);
    float Z2o = __shfl_xor(Z2, 1, 32);
    float mm  = fmaxf(m2, m2o);
    Z2 = Z2 * __expf(m2 - mm) + Z2o * __expf(m2o - mm);
    m2 = mm;
  }
  // pass B1: Z1 = sum exp(s - m1)
  float Z1 = 0.0f;
  for (int i = 0; i < 256; ++i) Z1 += __expf(Srow[i] - m1);
  Z1 += __shfl_xor(Z1, 1, 32);
  const float lse1 = m1 + __logf(Z1);
  const float lse2 = m2 + __logf(Z2);

  // pass B2: p, avg_probs, sum p*logp, enc (overwrites s in LDS)
  float kacc = 0.0f;
  for (int i = 0; i < 256; ++i) {
    int kk = (i + 2 * mloc) & 255;          // skew: no AVG atomic contention
    int k  = half * 256 + kk;
    float s  = S[mloc * S_STR + k];
    float lp = s - lse1;
    float p  = __expf(lp);
    atomicAdd(&AVG[k], p);
    kacc += p * lp;
    float g   = gumbel_from_index((unsigned)(token * 512 + k));
    float enc = __expf(2.0f * (s + g) - lse2);
    S[mloc * S_STR + k] = enc;
  }
  atomicAdd(&RED[0], kacc);
  __syncthreads();

  // flush block accumulators to global workspace
  {
    int k = tid * 4;
    atomicAdd(&ws[k + 0], AVG[k + 0]);
    atomicAdd(&ws[k + 1], AVG[k + 1]);
    atomicAdd(&ws[k + 2], AVG[k + 2]);
    atomicAdd(&ws[k + 3], AVG[k + 3]);
  }
  if (tid == 0) atomicAdd(&ws[512], RED[0]);

  // ================= GEMM2: zq = enc @ cb  (all from LDS) ==================
  const float* Sa = S + (wave * 16 + lo16) * S_STR + 2 * hi;
  for (int nt = 0; nt < 4; ++nt) {
    const float* Bb = CB + (2 * hi) * CB_STR + nt * 16 + lo16;
    v8f c0 = {}, c1 = {};
#pragma unroll 4
    for (int k0 = 0; k0 < 128; k0 += 2) {
      v2f av0 = *(const v2f*)(Sa + 4 * k0);
      v2f av1 = *(const v2f*)(Sa + 4 * k0 + 4);
      v2f bv0, bv1;
      bv0.x = Bb[(4 * k0 + 0) * CB_STR];
      bv0.y = Bb[(4 * k0 + 1) * CB_STR];
      bv1.x = Bb[(4 * k0 + 4) * CB_STR];
      bv1.y = Bb[(4 * k0 + 5) * CB_STR];
      c0 = wmma_f32_4(av0, bv0, c0);
      c1 = wmma_f32_4(av1, bv1, c1);
    }
    const int d  = nt * 16 + lo16;
    const int rb = wave * 16 + 8 * hi;
#pragma unroll
    for (int vg = 0; vg < 8; ++vg) ZQ[d * ZQ_STR + rb + vg] = c0[vg] + c1[vg];
  }
  __syncthreads();

  // coalesced store: out[b][d][pix], 64 contiguous tokens per block
  {
    const int d = tid >> 1;
    const int mstart = (tid & 1) * 32;
    const int b64s = (blockBase >> 10) * 64;
    const int pix0 = blockBase & 1023;
    float* orow = out + (((b64s + d) << 10) + pix0 + mstart);
    const float* zrow = ZQ + d * ZQ_STR + mstart;
#pragma unroll
    for (int mm = 0; mm < 32; mm += 4)
      *(float4*)(orow + mm) = *(const float4*)(zrow + mm);
  }
}

// ---- zero workspace accumulators -------------------------------------------
extern "C" __global__ void vq_prep_kernel(float* __restrict__ ws) {
  for (int i = threadIdx.x; i < 514; i += 256) ws[i] = 0.0f;
}

// ---- kld_continuous: sum (z - zq)^2 ----------------------------------------
extern "C" __global__ void __launch_bounds__(256)
vq_kldc_kernel(const float* __restrict__ z, const float* __restrict__ out,
               float* __restrict__ ws) {
  __shared__ float red[8];
  const int i0 = (blockIdx.x * 256 + threadIdx.x) * 4;
  float4 zv = *(const float4*)(z + i0);
  float4 ov = *(const float4*)(out + i0);
  float d0 = zv.x - ov.x, d1 = zv.y - ov.y, d2 = zv.z - ov.z, d3 = zv.w - ov.w;
  float s = d0 * d0 + d1 * d1 + d2 * d2 + d3 * d3;
#pragma unroll
  for (int off = 16; off > 0; off >>= 1) s += __shfl_xor(s, off, 32);
  if ((threadIdx.x & 31) == 0) red[threadIdx.x >> 5] = s;
  __syncthreads();
  if (threadIdx.x == 0) {
    float t = 0.0f;
#pragma unroll
    for (int q = 0; q < 8; ++q) t += red[q];
    atomicAdd(&ws[513], t);
  }
}

// ---- loss + perplexity scalars ---------------------------------------------
extern "C" __global__ void __launch_bounds__(512)
vq_final_kernel(const float* __restrict__ var_q, const float* __restrict__ ws,
                float* __restrict__ out2) {
  __shared__ float red[16];
  const int t = threadIdx.x;               // 512 threads, one per code
  float a = ws[t] * (1.0f / 65536.0f);     // avg_probs
  float term = a * __logf(a + 1e-7f);
#pragma unroll
  for (int off = 16; off > 0; off >>= 1) term += __shfl_xor(term, off, 32);
  if ((t & 31) == 0) red[t >> 5] = term;
  __syncthreads();
  if (t == 0) {
    float ssum = 0.0f;
#pragma unroll
    for (int q = 0; q < 16; ++q) ssum += red[q];
    float w = 0.5f / fmaxf(var_q[0], 1e-10f);
    float loss = ws[512] * (1.0f / 64.0f) + ws[513] * w * (1.0f / 64.0f);
    out2[0] = loss;            // kld_discrete + kld_continuous
    out2[1] = __expf(-ssum);   // perplexity
  }
}

extern "C" void kernel_launch(void* const* d_in, const int* in_sizes, int n_in,
                              void* d_out, int out_size, void* d_ws,
                              size_t ws_size, hipStream_t stream) {
  const float* z     = (const float*)d_in[0];
  const float* var_q = (const float*)d_in[1];
  const float* cb    = (const float*)d_in[2];
  float* out = (float*)d_out;
  float* ws  = (float*)d_ws;

  const size_t smem = SM_FLOATS * sizeof(float);   // ~289 KB of 320 KB/WGP
  hipFuncSetAttribute(reinterpret_cast<const void*>(vq_main_kernel),
                      hipFuncAttributeMaxDynamicSharedMemorySize, (int)smem);

  vq_prep_kernel<<<1, 256, 0, stream>>>(ws);
  vq_main_kernel<<<1024, 128, smem, stream>>>(z, var_q, cb, out, ws);
  vq_kldc_kernel<<<4096, 256, 0, stream>>>(z, out, ws);
  vq_final_kernel<<<1, 512, 0, stream>>>(var_q, ws, out + 4194304);
}